// NoiseEstimation_17463337025964
// MI455X (gfx1250) — compile-verified
//
#include <hip/hip_runtime.h>

// Problem constants from the reference: (B=16, C=1, H=W=512) fp32 in,
// (B, 3, H, W) fp32 out.
static constexpr int HH = 512;
static constexpr int WW = 512;
static constexpr int TILE = 32;
static constexpr int IMG_DIM = TILE + 4;   // 36: 5x5 window / theta+sobel halo
static constexpr int IMG_STR = IMG_DIM + 1; // 37: LDS row stride
static constexpr int U_DIM = TILE + 2;     // 34: theta ring (halo 1)
static constexpr int U_STR = U_DIM + 1;    // 35

__global__ __launch_bounds__(256) void
noise_estimation_kernel(const float* __restrict__ img, float* __restrict__ out)
{
    __shared__ float  sImg[IMG_DIM * IMG_STR];  // raw image tile (zero-padded)
    __shared__ float2 sU[U_DIM * U_STR];        // unit gradient dir (cosθ, sinθ)
    __shared__ float  sG[U_DIM * U_STR];        // gradient magnitude

    const int tid = threadIdx.x;
    const int ox  = blockIdx.x * TILE;
    const int oy  = blockIdx.y * TILE;
    const int b   = blockIdx.z;

    const float* __restrict__ src = img + (size_t)b * HH * WW;

    // ---- Phase 1: async global -> LDS tile load (CDNA5 ASYNC path) ----
    // 36*36 = 1296 slots, 6 rounds of 256 lanes. Indices are clamped (not
    // predicated) so EXEC stays all-ones for the async-VMEM instruction;
    // duplicate slot writes carry identical data. Out-of-image slots are
    // loaded from a clamped address and overwritten with 0 afterwards
    // (zero-padding semantics for Sobel and the 5x5 variance window).
#pragma unroll
    for (int k = 0; k < 6; ++k) {
        int idx = tid + k * 256;
        if (idx > IMG_DIM * IMG_DIM - 1) idx = IMG_DIM * IMG_DIM - 1;
        const int ty = idx / IMG_DIM;
        const int tx = idx - ty * IMG_DIM;
        const int gy = oy + ty - 2;
        const int gx = ox + tx - 2;
        const int gyc = min(max(gy, 0), HH - 1);
        const int gxc = min(max(gx, 0), WW - 1);
        const unsigned goff   = (unsigned)((gyc * WW + gxc) * 4);
        const unsigned ldsoff =
            (unsigned)(uintptr_t)(&sImg[ty * IMG_STR + tx]);
        asm volatile("global_load_async_to_lds_b32 %0, %1, %2"
                     :
                     : "v"(ldsoff), "v"(goff), "s"(src)
                     : "memory");
    }
    asm volatile("s_wait_asynccnt 0" ::: "memory");
    __syncthreads();

    // Zero-fix out-of-image slots (zero padding).
    for (int idx = tid; idx < IMG_DIM * IMG_DIM; idx += 256) {
        const int ty = idx / IMG_DIM;
        const int tx = idx - ty * IMG_DIM;
        const int gy = oy + ty - 2;
        const int gx = ox + tx - 2;
        if ((unsigned)gy >= (unsigned)HH || (unsigned)gx >= (unsigned)WW)
            sImg[ty * IMG_STR + tx] = 0.0f;
    }
    __syncthreads();

    // ---- Phase 2: Sobel -> (G, unit direction) on the 34x34 ring ----
    // Coordinates clamped to the image => replicate padding for theta.
    for (int idx = tid; idx < U_DIM * U_DIM; idx += 256) {
        const int ty = idx / U_DIM;
        const int tx = idx - ty * U_DIM;
        const int gy = oy + ty - 1;
        const int gx = ox + tx - 1;
        const int gyc = min(max(gy, 0), HH - 1);
        const int gxc = min(max(gx, 0), WW - 1);
        const int iy = gyc - oy + 2;   // in [1, 34]
        const int ix = gxc - ox + 2;

        const float* p = &sImg[(iy - 1) * IMG_STR + (ix - 1)];
        const float i00 = p[0],           i01 = p[1],           i02 = p[2];
        const float i10 = p[IMG_STR],     i11 = p[IMG_STR + 1], i12 = p[IMG_STR + 2];
        const float i20 = p[2 * IMG_STR], i21 = p[2 * IMG_STR + 1], i22 = p[2 * IMG_STR + 2];
        (void)i11;

        // cross-correlation with kx = [[-1,0,1],[-2,0,2],[-1,0,1]] and kx^T
        const float gxs = (i02 + 2.0f * i12 + i22) - (i00 + 2.0f * i10 + i20);
        const float gys = (i20 + 2.0f * i21 + i22) - (i00 + 2.0f * i01 + i02);
        const float g   = sqrtf(gxs * gxs + gys * gys);
        const float inv = (g > 0.0f) ? (1.0f / g) : 0.0f;
        float2 u;
        u.x = (g > 0.0f) ? (gxs * inv) : 1.0f;  // cos(atan2(0,0)) = 1
        u.y = gys * inv;                         // sin(atan2(0,0)) = 0
        sU[ty * U_STR + tx] = u;
        sG[ty * U_STR + tx] = g;
    }
    __syncthreads();

    // ---- Phase 3: per-pixel G, C, V; coalesced stores ----
    const size_t HWsz  = (size_t)HH * WW;
    float* __restrict__ dst = out + (size_t)b * 3 * HWsz;

    for (int idx = tid; idx < TILE * TILE; idx += 256) {
        const int ly = idx >> 5;          // TILE == 32
        const int lx = idx & 31;
        const int uy = ly + 1;
        const int ux = lx + 1;

        // C: mean over 8 neighbors of cos(theta - theta_n) = u . u_n
        const float2 uc = sU[uy * U_STR + ux];
        float csum = 0.0f;
#pragma unroll
        for (int dy = -1; dy <= 1; ++dy) {
#pragma unroll
            for (int dx = -1; dx <= 1; ++dx) {
                if (dy == 0 && dx == 0) continue;
                const float2 un = sU[(uy + dy) * U_STR + (ux + dx)];
                csum += uc.x * un.x + uc.y * un.y;
            }
        }
        const float C = csum * 0.125f;
        const float G = sG[uy * U_STR + ux];

        // V: variance of the zero-padded 5x5 window
        const int iy = ly + 2;
        const int ix = lx + 2;
        float s1 = 0.0f, s2 = 0.0f;
#pragma unroll
        for (int dy = -2; dy <= 2; ++dy) {
#pragma unroll
            for (int dx = -2; dx <= 2; ++dx) {
                const float v = sImg[(iy + dy) * IMG_STR + (ix + dx)];
                s1 += v;
                s2 = fmaf(v, v, s2);
            }
        }
        const float mu = s1 * (1.0f / 25.0f);
        const float V  = fmaf(-mu, mu, s2 * (1.0f / 25.0f));

        const size_t o = (size_t)(oy + ly) * WW + (ox + lx);
        dst[o]             = G;
        dst[o + HWsz]      = C;
        dst[o + 2 * HWsz]  = V;
    }
}

extern "C" void kernel_launch(void* const* d_in, const int* in_sizes, int n_in,
                              void* d_out, int out_size, void* d_ws, size_t ws_size,
                              hipStream_t stream)
{
    (void)n_in; (void)out_size; (void)d_ws; (void)ws_size;
    const float* img = (const float*)d_in[0];   // (B,1,512,512) fp32
    // d_in[1], d_in[2]: Sobel weights — constant, folded into the kernel.
    float* out = (float*)d_out;                 // (B,3,512,512) fp32

    const int B = in_sizes[0] / (HH * WW);
    dim3 grid(WW / TILE, HH / TILE, B);
    noise_estimation_kernel<<<grid, 256, 0, stream>>>(img, out);
}